// edr_loss_78271484002674
// MI455X (gfx1250) — compile-verified
//
#include <hip/hip_runtime.h>
#include <hip/hip_bf16.h>

// ---------------------------------------------------------------------------
// EDR loss pipeline for MI455X (gfx1250, wave32, WMMA f32 16x16x4)
//
// Factorization: irfft(96000) = outer-DFT375 ( twiddle ( inner-DFT256 ) )
//   k = k1 + 375*k2, n = 256*n1 + n2
// Stage2 GEMM: [375 x 512]([Yr|Yi]) x [512 x 512] block-DFT   (per signal-batch)
// StageT     : twiddle * transpose -> [256 x 752] = [Zr | 0 | Zi | 0]
// Stage3 GEMM: [256 x 752] x [752 x 384]([C;-S])  -> x (real part), transposed store
// Stage4 GEMM: frames [374 x 512] (lda=hop=256, 50% overlap) x [512 x 640]
//              (Hann folded into B: [w*cos | -w*sin | 0-pad])
// Stage5     : reverse-cumsum power -> 10log10 -> relative L1 reduction
//
// GEMM: 8 waves/block, each wave a 2x2 grid of 16x16 WMMA tiles (block 64x128).
// B (shared DFT matrix) staged to LDS in 32x128 chunks via
// GLOBAL_LOAD_ASYNC_TO_LDS_B128 (ASYNCcnt + s_wait_asynccnt), ds fallback.
// ---------------------------------------------------------------------------

typedef float v2f __attribute__((ext_vector_type(2)));
typedef float v8f __attribute__((ext_vector_type(8)));
typedef int   v4i_vec __attribute__((vector_size(16)));   // matches builtin param

#define TWO_PI 6.28318530717958647692f
#define NFULL  96000
#define N1D    375     // outer DFT length
#define N2D    256     // inner DFT length
#define NSB    256     // 2 signals * 128 batch
#define WINS   512
#define HOP    256
#define TFR    374     // STFT frames
#define NFREQ  257
#define S4LD   514     // [re(257) | im(257)] per frame row
#define KC     32      // K-chunk staged in LDS
#define NB     128     // block column-tile width

#if defined(__has_builtin)
#if __has_builtin(__builtin_amdgcn_global_load_async_to_lds_b128) && \
    __has_builtin(__builtin_amdgcn_s_wait_asynccnt)
#define USE_ASYNC_LDS 1
#endif
#endif
#ifndef USE_ASYNC_LDS
#define USE_ASYNC_LDS 0
#endif

// ---------------- generic fp32 WMMA GEMM:  C = A(MxK,lda) * B(KxN=ldb) -----
__global__ __launch_bounds__(256)
void wmma_gemm_f32(const float* __restrict__ A, const float* __restrict__ B,
                   float* __restrict__ C,
                   int M, int Nstore, int K, int lda, int ldb,
                   long long astride, long long cstride, int crs, int ccs)
{
  __shared__ float bsh[KC * NB];   // 16 KB B-chunk

  const int sb = blockIdx.z;
  const float* Ab = A + (long long)sb * astride;
  float*       Cb = C + (long long)sb * cstride;

  const int tid  = threadIdx.x;
  const int lane = tid & 31;
  const int wave = tid >> 5;
  const int l15  = lane & 15;
  const int hi   = lane >> 4;      // 0: K pair (0,1)   1: K pair (2,3)
  const int koff = hi * 2;

  const int wm  = blockIdx.y * 64  + (wave >> 2) * 32;   // wave 2x2 tile origin
  const int wn  = blockIdx.x * NB  + (wave & 3) * 32;
  const int bn0 = blockIdx.x * NB;                       // block B column base
  const int wc  = (wave & 3) * 32;                       // wave column in LDS chunk

  int ar0 = wm + l15;      if (ar0 >= M) ar0 = M - 1;    // clamp; stores guarded
  int ar1 = wm + 16 + l15; if (ar1 >= M) ar1 = M - 1;
  const float* pA0 = Ab + (long long)ar0 * lda;
  const float* pA1 = Ab + (long long)ar1 * lda;

  v8f acc00 = {}, acc01 = {}, acc10 = {}, acc11 = {};

  for (int kk = 0; kk < K; kk += KC) {
    const int kc = (K - kk < KC) ? (K - kk) : KC;        // uniform across block

    // ---- stage B chunk [kc x 128] into LDS (async-to-LDS when available) --
    for (int q = tid; q < kc * (NB / 4); q += 256) {     // kc*32 float4s, uniform trips
      const int r = q >> 5;
      const int c = (q & 31) * 4;
      const float* gsrc = &B[(long long)(kk + r) * ldb + bn0 + c];
#if USE_ASYNC_LDS
      __builtin_amdgcn_global_load_async_to_lds_b128(
          (__attribute__((address_space(1))) v4i_vec*)(unsigned long long)gsrc,
          (__attribute__((address_space(3))) v4i_vec*)(unsigned int)(unsigned long long)&bsh[r * NB + c],
          0, 0);
#else
      *(float4*)&bsh[r * NB + c] = *(const float4*)gsrc;
#endif
    }
    // prefetch next chunk of B toward L2 while this one computes
    {
      const int pr = tid >> 3, pc = (tid & 7) * 16;
      if (kk + KC + pr < K)
        __builtin_prefetch(&B[(long long)(kk + KC + pr) * ldb + bn0 + pc], 0, 1);
    }
#if USE_ASYNC_LDS
    __builtin_amdgcn_s_wait_asynccnt(0);
#endif
    __syncthreads();

    // ---- compute: 4 WMMA per K-step of 4, fragments from LDS/global -------
    for (int k0 = 0; k0 < kc; k0 += 4) {
      const float4 a4_0 = *(const float4*)(pA0 + kk + k0);
      const float4 a4_1 = *(const float4*)(pA1 + kk + k0);
      v2f a0, a1;
      a0[0] = hi ? a4_0.z : a4_0.x;  a0[1] = hi ? a4_0.w : a4_0.y;
      a1[0] = hi ? a4_1.z : a4_1.x;  a1[1] = hi ? a4_1.w : a4_1.y;
      const float* brow0 = &bsh[(k0 + koff)     * NB + wc + l15];
      const float* brow1 = &bsh[(k0 + koff + 1) * NB + wc + l15];
      v2f bA, bB;
      bA[0] = brow0[0];  bA[1] = brow1[0];
      bB[0] = brow0[16]; bB[1] = brow1[16];
      acc00 = __builtin_amdgcn_wmma_f32_16x16x4_f32(false, a0, false, bA, (short)0, acc00, false, false);
      acc01 = __builtin_amdgcn_wmma_f32_16x16x4_f32(false, a0, false, bB, (short)0, acc01, false, false);
      acc10 = __builtin_amdgcn_wmma_f32_16x16x4_f32(false, a1, false, bA, (short)0, acc10, false, false);
      acc11 = __builtin_amdgcn_wmma_f32_16x16x4_f32(false, a1, false, bB, (short)0, acc11, false, false);
    }
    __syncthreads();
  }

  // ---- store 2x2 tiles; C/D layout: VGPR v -> row base + hi*8 + v ---------
  const int rb = hi * 8;
#pragma unroll
  for (int v = 0; v < 8; ++v) {
    const int r0 = wm + rb + v, r1 = wm + 16 + rb + v;
    const int c0 = wn + l15,    c1 = wn + 16 + l15;
    if (r0 < M && c0 < Nstore) Cb[(long long)r0 * crs + (long long)c0 * ccs] = acc00[v];
    if (r0 < M && c1 < Nstore) Cb[(long long)r0 * crs + (long long)c1 * ccs] = acc01[v];
    if (r1 < M && c0 < Nstore) Cb[(long long)r1 * crs + (long long)c0 * ccs] = acc10[v];
    if (r1 < M && c1 < Nstore) Cb[(long long)r1 * crs + (long long)c1 * ccs] = acc11[v];
  }
}

// ---------------- stage 1: Hermitian expand + gather into GEMM-A layout ----
__global__ void expand_spectrum(const float* __restrict__ tre, const float* __restrict__ tim,
                                const float* __restrict__ are, const float* __restrict__ aim,
                                float* __restrict__ A2)
{
  long long idx = (long long)blockIdx.x * blockDim.x + threadIdx.x;
  const long long total = (long long)NSB * N1D * N2D;
  if (idx >= total) return;
  const int k2 = (int)(idx % N2D);
  long long r  = idx / N2D;
  const int k1 = (int)(r % N1D);
  const int sb = (int)(r / N1D);
  const int s = sb >> 7, b = sb & 127;
  const float* RE = s ? are : tre;
  const float* IM = s ? aim : tim;
  const int k = k1 + N1D * k2;          // in [0, 96000)
  float re, im;
  if (k <= NFULL / 2) {                 // irfft crops to 48001 bins
    re = RE[(long long)b * NFULL + k];
    im = (k == 0 || k == NFULL / 2) ? 0.0f : IM[(long long)b * NFULL + k];
  } else {
    const int kk = NFULL - k;
    re =  RE[(long long)b * NFULL + kk];
    im = -IM[(long long)b * NFULL + kk];
  }
  const float inv = 1.0f / (float)NFULL;
  float* row = A2 + ((long long)sb * N1D + k1) * 512;
  row[k2]       = re * inv;
  row[256 + k2] = im * inv;
}

// ---------------- DFT matrix builders --------------------------------------
__global__ void build_b2(float* __restrict__ B2) {   // [512 x 512] block inverse-DFT256
  int idx = blockIdx.x * blockDim.x + threadIdx.x;
  if (idx >= 512 * 512) return;
  const int n = idx & 511, j = idx >> 9;
  const int k2 = j & 255, n2 = n & 255;
  const float th = (TWO_PI / 256.0f) * (float)((k2 * n2) & 255);
  float s, c; __sincosf(th, &s, &c);
  float v;
  if (j < 256) v = (n < 256) ?  c : s;   // [ Wr  Wi ]
  else         v = (n < 256) ? -s : c;   // [-Wi  Wr ]
  B2[idx] = v;
}

__global__ void build_b3(float* __restrict__ B3) {   // [752 x 384] = [C; pad; -S; pad]
  int idx = blockIdx.x * blockDim.x + threadIdx.x;
  if (idx >= 752 * 384) return;
  const int n1 = idx % 384, j = idx / 384;
  float v = 0.0f;
  if (n1 < N1D) {
    if (j < N1D) {
      const int m = (j * n1) % N1D;
      v = __cosf((TWO_PI / (float)N1D) * (float)m);
    } else if (j >= 376 && j < 751) {
      const int m = ((j - 376) * n1) % N1D;
      v = -__sinf((TWO_PI / (float)N1D) * (float)m);
    }
  }
  B3[idx] = v;
}

__global__ void build_b4(float* __restrict__ B4) {   // [512 x 640] Hann-folded rfft512
  int idx = blockIdx.x * blockDim.x + threadIdx.x;
  if (idx >= 512 * 640) return;
  const int c = idx % 640, j = idx / 640;
  const float w = 0.5f * (1.0f - __cosf((TWO_PI / (float)WINS) * (float)j));
  float v = 0.0f;
  if (c < NFREQ) {
    v =  w * __cosf((TWO_PI / (float)WINS) * (float)((j * c) & 511));
  } else if (c < 2 * NFREQ) {
    const int f = c - NFREQ;
    v = -w * __sinf((TWO_PI / (float)WINS) * (float)((j * f) & 511));
  }
  B4[idx] = v;
}

// ---------------- twiddle + transpose: C2[sb][k1][512] -> A3[sb][n2][752] --
__global__ void twiddle_transpose(const float* __restrict__ C2, float* __restrict__ A3)
{
  long long idx = (long long)blockIdx.x * blockDim.x + threadIdx.x;
  const long long total = (long long)NSB * N2D * 376;
  if (idx >= total) return;
  const int k1 = (int)(idx % 376);
  long long r  = idx / 376;
  const int n2 = (int)(r % N2D);
  const int sb = (int)(r / N2D);
  float* row = A3 + ((long long)sb * N2D + n2) * 752;
  if (k1 == 375) { row[375] = 0.0f; row[751] = 0.0f; return; }
  const float* c2 = C2 + ((long long)sb * N1D + k1) * 512;
  const float zr = c2[n2], zi = c2[256 + n2];
  const float th = (TWO_PI / (float)NFULL) * (float)(k1 * n2);  // k1*n2 < 96000: exact
  float s, c; __sincosf(th, &s, &c);
  row[k1]       = zr * c - zi * s;   // Zr'
  row[376 + k1] = zr * s + zi * c;   // Zi'
}

// ---------------- stage 5: EDR (reverse cumsum) + dB + relative L1 --------
__global__ __launch_bounds__(256)
void edr_loss_reduce(const float* __restrict__ S4, float* __restrict__ acc)
{
  __shared__ float sn[256];
  __shared__ float sd[256];
  const int idx = blockIdx.x * 256 + threadIdx.x;
  float num = 0.0f, den = 0.0f;
  if (idx < 128 * NFREQ) {
    const int b = idx / NFREQ, f = idx % NFREQ;
    const float* St = S4 + (long long)b         * (TFR * S4LD);
    const float* Sa = S4 + (long long)(128 + b) * (TFR * S4LD);
    float at = 0.0f, aa = 0.0f;
    for (int t = TFR - 1; t >= 0; --t) {
      float re = St[t * S4LD + f], im = St[t * S4LD + NFREQ + f];
      at += re * re + im * im;
      re = Sa[t * S4LD + f]; im = Sa[t * S4LD + NFREQ + f];
      aa += re * re + im * im;
      const float dt = 10.0f * log10f(at);
      const float da = 10.0f * log10f(aa);
      num += fabsf(dt - da);
      den += fabsf(dt);
    }
  }
  sn[threadIdx.x] = num; sd[threadIdx.x] = den;
  __syncthreads();
  for (int off = 128; off > 0; off >>= 1) {
    if (threadIdx.x < off) {
      sn[threadIdx.x] += sn[threadIdx.x + off];
      sd[threadIdx.x] += sd[threadIdx.x + off];
    }
    __syncthreads();
  }
  if (threadIdx.x == 0) { atomicAdd(&acc[0], sn[0]); atomicAdd(&acc[1], sd[0]); }
}

__global__ void init_acc(float* acc) { acc[0] = 0.0f; acc[1] = 0.0f; }
__global__ void finalize(const float* acc, float* out) { out[0] = acc[0] / acc[1]; }

// ---------------------------------------------------------------------------
static inline unsigned cdiv(long long a, long long b) { return (unsigned)((a + b - 1) / b); }

extern "C" void kernel_launch(void* const* d_in, const int* in_sizes, int n_in,
                              void* d_out, int out_size, void* d_ws, size_t ws_size,
                              hipStream_t stream)
{
  const float* tre = (const float*)d_in[0];
  const float* tim = (const float*)d_in[1];
  const float* are = (const float*)d_in[2];
  const float* aim = (const float*)d_in[3];
  float* out = (float*)d_out;
  float* ws  = (float*)d_ws;

  // workspace layout (floats); offsets multiples of 4 for 16B alignment
  float* B2   = ws;                    // 262,144
  float* B3   = B2 + 262144;           // 288,768
  float* B4   = B3 + 288768;           // 327,680  (512 x 640)
  float* acc  = B4 + 327680;           // 16
  float* buf1 = acc + 16;              // 49,283,072 : A2 [256*375*512] then A3 [256*256*752]
  float* buf2 = buf1 + 49283072LL;     // 49,212,416 : C2 [256*375*512] then S4 [256*374*514]
  float* X    = buf2 + 49212416LL;     // 24,576,000 : time-domain x [256*96000]

  const dim3 blk(256);

  // DFT matrices
  build_b2<<<cdiv(512LL * 512, 256), blk, 0, stream>>>(B2);
  build_b3<<<cdiv(752LL * 384, 256), blk, 0, stream>>>(B3);
  build_b4<<<cdiv(512LL * 640, 256), blk, 0, stream>>>(B4);

  // stage 1: Hermitian expand + gather -> A2 (buf1)
  expand_spectrum<<<cdiv((long long)NSB * N1D * N2D, 256), blk, 0, stream>>>(
      tre, tim, are, aim, buf1);

  // stage 2: inner inverse DFT256 as GEMM: [375x512] x [512x512] -> C2 (buf2)
  wmma_gemm_f32<<<dim3(4, 6, NSB), blk, 0, stream>>>(
      buf1, B2, buf2,
      /*M=*/N1D, /*Nstore=*/512, /*K=*/512, /*lda=*/512, /*ldb=*/512,
      (long long)N1D * 512, (long long)N1D * 512, /*crs=*/512, /*ccs=*/1);

  // stage T: twiddle + transpose -> A3 (buf1, A2 dead)
  twiddle_transpose<<<cdiv((long long)NSB * N2D * 376, 256), blk, 0, stream>>>(buf2, buf1);

  // stage 3: outer DFT375 (real part) as GEMM: [256x752] x [752x384] -> x
  // transposed store: x[256*n1 + n2]  (crs=1, ccs=256)
  wmma_gemm_f32<<<dim3(3, 4, NSB), blk, 0, stream>>>(
      buf1, B3, X,
      /*M=*/N2D, /*Nstore=*/N1D, /*K=*/752, /*lda=*/752, /*ldb=*/384,
      (long long)N2D * 752, (long long)NFULL, /*crs=*/1, /*ccs=*/256);

  // stage 4: STFT as GEMM with overlapping frames (lda = hop = 256),
  // Hann folded into B4: [374x512] x [512x640] -> S4 (buf2, C2 dead)
  wmma_gemm_f32<<<dim3(5, 6, NSB), blk, 0, stream>>>(
      X, B4, buf2,
      /*M=*/TFR, /*Nstore=*/S4LD, /*K=*/WINS, /*lda=*/HOP, /*ldb=*/640,
      (long long)NFULL, (long long)TFR * S4LD, /*crs=*/S4LD, /*ccs=*/1);

  // stage 5: EDR reverse-cumsum, dB, relative-L1 reduction, final divide
  init_acc<<<1, 1, 0, stream>>>(acc);
  edr_loss_reduce<<<cdiv(128LL * NFREQ, 256), blk, 0, stream>>>(buf2, acc);
  finalize<<<1, 1, 0, stream>>>(acc, out);
}